// PatchEmbeder_18811956756829
// MI455X (gfx1250) — compile-verified
//
#include <hip/hip_runtime.h>

typedef float v2f __attribute__((ext_vector_type(2)));
typedef float v4f __attribute__((ext_vector_type(4)));
typedef float v8f __attribute__((ext_vector_type(8)));
typedef int   v4i __attribute__((ext_vector_type(4)));

// D = A(16x4,f32) * B(4x16,f32) + C(16x16,f32)
#define WMMA_F32(a, b, c) \
  __builtin_amdgcn_wmma_f32_16x16x4_f32(false, (a), false, (b), (short)0, (c), false, false)

#if defined(__has_builtin)
#if __has_builtin(__builtin_amdgcn_global_load_async_to_lds_b128)
#define HAVE_ASYNC_LDS 1
#endif
#endif
#ifndef HAVE_ASYNC_LDS
#define HAVE_ASYNC_LDS 0
#endif

constexpr int Bsz = 128;          // batch
constexpr int Cch = 3;            // channels
constexpr int Pn  = 196;          // patches per image
constexpr int HPx = 16, WPx = 16; // patch h/w
constexpr int K1  = Cch * HPx * WPx; // 768
constexpr int N1  = 32;
constexpr int N2  = 64;
constexpr int E   = 768;
constexpr int ROWS  = Bsz * Pn;   // 25088
constexpr int WAVES = 8;          // waves per block (256 threads, wave32)
constexpr int CHAN_STRIDE = Pn * HPx * WPx; // 50176 floats between channels

// ---- shared-memory pool layout (floats) ----
constexpr int C1_STRIDE = N1 + 1;              // 33 (bank-conflict pad)
constexpr int C2_STRIDE = N2 + 1;              // 65
constexpr int C1_WAVE   = 16 * C1_STRIDE;      // per-wave C1 staging
constexpr int C2_WAVE   = 16 * C2_STRIDE;      // per-wave C2 staging
constexpr int C2_OFF    = WAVES * C1_WAVE;     // 4224
constexpr int POOL_FLOATS = C2_OFF + WAVES * C2_WAVE; // 12544 floats = 49 KB
// W3 chunk staging reuses the pool after GEMM2 (A3 frags live in registers)
constexpr int CHUNK_NT   = 8;                  // N-tiles per staged W3 chunk
constexpr int CHUNK_COLS = CHUNK_NT * 16;      // 128
constexpr int W3S_STRIDE = 136;                // pad so lane groups hit disjoint banks
static_assert(N2 * W3S_STRIDE <= POOL_FLOATS, "W3 chunk must fit in pool");

__global__ __launch_bounds__(256) void patch_embed_kernel(
    const float* __restrict__ in,  const float* __restrict__ W1, const float* __restrict__ b1,
    const float* __restrict__ W2,  const float* __restrict__ b2,
    const float* __restrict__ W3,  const float* __restrict__ b3,
    const float* __restrict__ pos, float* __restrict__ out)
{
  __shared__ __align__(16) float smem[POOL_FLOATS];

  const int wave = threadIdx.x >> 5;
  const int lane = threadIdx.x & 31;
  const int g    = lane >> 4;   // lane group (selects K pair / M half)
  const int ln   = lane & 15;
  const int tile = blockIdx.x * WAVES + wave;   // 0..1567
  const int row0 = tile * 16;

  float* const c1s = smem + wave * C1_WAVE;            // [16][33]
  float* const c2s = smem + C2_OFF + wave * C2_WAVE;   // [16][65]

  // ---- A addressing for GEMM1: this lane owns row M = ln of the tile ----
  const int rA = row0 + ln;                 // global (b,p) row
  const int bA = rA / Pn;
  const int pA = rA - bA * Pn;
  const long rowBase = ((long)bA * (Cch * Pn) + pA) * (HPx * WPx);

  // ---- GEMM1: [16x768] @ [768x32] + b1 ----
  v8f acc0 = {}; v8f acc1 = {};
  for (int kb = 0; kb < K1 / 4; ++kb) {
    const int k0  = 4 * kb + 2 * g;         // lane-group's first K of this block
    const int c   = k0 >> 8;                // channel
    const int off = k0 & 255;               // offset within channel slab
    v2f a = *(const v2f*)(in + rowBase + (long)c * CHAN_STRIDE + off);
    v2f bb0, bb1;
    bb0.x = W1[k0 * N1 + ln];        bb0.y = W1[(k0 + 1) * N1 + ln];
    bb1.x = W1[k0 * N1 + 16 + ln];   bb1.y = W1[(k0 + 1) * N1 + 16 + ln];
    acc0 = WMMA_F32(a, bb0, acc0);
    acc1 = WMMA_F32(a, bb1, acc1);
  }
  {
    const float bv0 = b1[ln], bv1 = b1[16 + ln];
    #pragma unroll
    for (int v = 0; v < 8; ++v) {
      c1s[(v + 8 * g) * C1_STRIDE + ln]      = acc0[v] + bv0;   // (M, N) of C1
      c1s[(v + 8 * g) * C1_STRIDE + 16 + ln] = acc1[v] + bv1;
    }
  }

  // ---- GEMM2: [16x32] @ [32x64] + b2  (per-wave LDS region, no barrier) ----
  v8f acc2[4] = {{}, {}, {}, {}};
  for (int kb = 0; kb < N1 / 4; ++kb) {
    const int k0 = 4 * kb + 2 * g;
    v2f a; a.x = c1s[ln * C1_STRIDE + k0]; a.y = c1s[ln * C1_STRIDE + k0 + 1];
    #pragma unroll
    for (int nt = 0; nt < 4; ++nt) {
      v2f bb;
      bb.x = W2[k0 * N2 + nt * 16 + ln];
      bb.y = W2[(k0 + 1) * N2 + nt * 16 + ln];
      acc2[nt] = WMMA_F32(a, bb, acc2[nt]);
    }
  }
  #pragma unroll
  for (int nt = 0; nt < 4; ++nt) {
    const float bv = b2[nt * 16 + ln];
    #pragma unroll
    for (int v = 0; v < 8; ++v)
      c2s[(v + 8 * g) * C2_STRIDE + nt * 16 + ln] = acc2[nt][v] + bv;
  }

  // ---- epilogue row offsets: out row = b*197 + p + 1, pos row = p + 1 ----
  int outOff[8], posOff[8];
  #pragma unroll
  for (int v = 0; v < 8; ++v) {
    const int r  = row0 + v + 8 * g;
    const int bb = r / Pn;
    const int pp = r - bb * Pn;
    outOff[v] = (bb * (Pn + 1) + pp + 1) * E;
    posOff[v] = (pp + 1) * E;
  }

  // ---- GEMM3 A-frags (16x64) held in registers ----
  v2f a3[N2 / 4];
  #pragma unroll
  for (int kb = 0; kb < N2 / 4; ++kb) {
    const int k0 = 4 * kb + 2 * g;
    a3[kb].x = c2s[ln * C2_STRIDE + k0];
    a3[kb].y = c2s[ln * C2_STRIDE + k0 + 1];
  }

  // ---- GEMM3: [16x64] @ [64x768] + b3 + pos_enc ----
  // W3 is staged through LDS per block (8x fewer L2 reads of W3).
  for (int ch = 0; ch < E / CHUNK_COLS; ++ch) {
    __syncthreads();                       // pool free / previous chunk consumed
    const int c0 = ch * CHUNK_COLS;
    // cooperative stage of W3[0:64, c0:c0+128] -> smem (row stride W3S_STRIDE)
    #pragma unroll
    for (int i = 0; i < 8; ++i) {
      const int q  = threadIdx.x + 256 * i;   // 0..2047 float4 slots
      const int r  = q >> 5;                  // row 0..63
      const int c4 = (q & 31) << 2;           // col 0..124 step 4
#if HAVE_ASYNC_LDS
      typedef __attribute__((address_space(1))) v4i* g4p;
      typedef __attribute__((address_space(3))) v4i* l4p;
      __builtin_amdgcn_global_load_async_to_lds_b128(
          (g4p)(W3 + (long)r * E + c0 + c4),
          (l4p)(smem + r * W3S_STRIDE + c4), 0, 0);
#else
      v4f t = *(const v4f*)(W3 + (long)r * E + c0 + c4);
      *(v4f*)(smem + r * W3S_STRIDE + c4) = t;
#endif
    }
#if HAVE_ASYNC_LDS
#if __has_builtin(__builtin_amdgcn_s_wait_asynccnt)
    __builtin_amdgcn_s_wait_asynccnt(0);
#else
    asm volatile("s_wait_asynccnt 0x0" ::: "memory");
#endif
#endif
    __syncthreads();                       // chunk visible to all waves

    for (int ntl = 0; ntl < CHUNK_NT; ++ntl) {
      const int n  = c0 + ntl * 16 + ln;   // global output column
      const int nl = ntl * 16 + ln;        // column within chunk
      v8f acc3 = {};
      #pragma unroll
      for (int kb = 0; kb < N2 / 4; ++kb) {
        const int k0 = 4 * kb + 2 * g;
        v2f bb;
        bb.x = smem[k0 * W3S_STRIDE + nl];
        bb.y = smem[(k0 + 1) * W3S_STRIDE + nl];
        acc3 = WMMA_F32(a3[kb], bb, acc3);
      }
      const float bias = b3[n];
      #pragma unroll
      for (int v = 0; v < 8; ++v)
        out[outOff[v] + n] = acc3[v] + bias + pos[posOff[v] + n];
    }
  }
}

// out[b, 0, :] = cls_token + pos_enc[0]
__global__ __launch_bounds__(256) void cls_row_kernel(
    const float* __restrict__ cls, const float* __restrict__ pos, float* __restrict__ out)
{
  const int idx = blockIdx.x * 256 + threadIdx.x;   // 0 .. B*E-1
  const int b = idx / E;
  const int e = idx - b * E;
  out[(long)b * (Pn + 1) * E + e] = cls[e] + pos[e];
}

extern "C" void kernel_launch(void* const* d_in, const int* in_sizes, int n_in,
                              void* d_out, int out_size, void* d_ws, size_t ws_size,
                              hipStream_t stream) {
  const float* in  = (const float*)d_in[0];
  const float* W1  = (const float*)d_in[1];
  const float* b1  = (const float*)d_in[2];
  const float* W2  = (const float*)d_in[3];
  const float* b2  = (const float*)d_in[4];
  const float* W3  = (const float*)d_in[5];
  const float* b3  = (const float*)d_in[6];
  const float* cls = (const float*)d_in[7];
  const float* pos = (const float*)d_in[8];
  float* out = (float*)d_out;

  patch_embed_kernel<<<ROWS / (16 * WAVES), 256, 0, stream>>>(
      in, W1, b1, W2, b2, W3, b3, pos, out);
  cls_row_kernel<<<(Bsz * E) / 256, 256, 0, stream>>>(cls, pos, out);
}